// Centeral_Difference_Loss_3891240370730
// MI455X (gfx1250) — compile-verified
//
#include <hip/hip_runtime.h>
#include <hip/hip_bf16.h>

typedef __attribute__((ext_vector_type(2))) float v2f;
typedef __attribute__((ext_vector_type(8))) float v8f;

// Problem dims (fixed by reference): x,y are [B=2, C=1, D=32, H=128, W=128] f32.
#define DIM_D 32
#define DIM_H 128
#define DIM_W 128
#define BATCH_STRIDE (DIM_D * DIM_H * DIM_W)   // 524288
#define TOTAL_TERMS  28311552.0f               // 27 * 2 * 32 * 128 * 128

// Tile: 4(d) x 8(h) x 32(w) voxels per block; halo +1 each side.
#define TD 4
#define TH 8
#define TW 32
#define HD (TD + 2)      // 6
#define HH (TH + 2)      // 10
#define HW (TW + 2)      // 34
#define HSZ (HD * HH * HW)   // 2040 floats
#define SH HW            // lds h-stride = 34
#define SD (HH * HW)     // lds d-stride = 340

// Full wave32 reduction using V_WMMA_F32_16X16X4_F32.
// A[m][0] = p_m (lanes 0-15, VGPR0), A[m][2] = p_{m+16} (lanes 16-31, VGPR0),
// A[m][1] = A[m][3] = 0 (VGPR1). B = ones(4x16).
// D[m][n] = p_m + p_{m+16} for every column n. Sum of the 8 C/D VGPRs in a lane
// covers half the M range; one xor-16 shuffle completes the total in all lanes.
__device__ __forceinline__ float wave_reduce_wmma(float v) {
    v2f a; a[0] = v;    a[1] = 0.0f;
    v2f b; b[0] = 1.0f; b[1] = 1.0f;
    v8f c = {};
    c = __builtin_amdgcn_wmma_f32_16x16x4_f32(
        /*neg_a=*/false, a, /*neg_b=*/false, b,
        /*c_mod=*/(short)0, c, /*reuse_a=*/false, /*reuse_b=*/false);
    float t = c[0] + c[1] + c[2] + c[3] + c[4] + c[5] + c[6] + c[7];
    t += __shfl_xor(t, 16, 32);
    return t;   // every lane now holds the 32-lane sum
}

__global__ void __launch_bounds__(256)
cdl_pass1(const float* __restrict__ x, const float* __restrict__ y,
          float* __restrict__ block_sums) {
    __shared__ float tile[HSZ];
    __shared__ float wsum[8];

    const int tid = threadIdx.x;
    const int bid = blockIdx.x;
    // 1024 blocks: wt fastest (4), then ht (16), dt (8), batch (2)
    const int wt = bid & 3;
    const int ht = (bid >> 2) & 15;
    const int dt = (bid >> 6) & 7;
    const int b  = bid >> 9;

    const int w0 = wt * TW;
    const int h0 = ht * TH;
    const int d0 = dt * TD;
    const float* xb = x + b * BATCH_STRIDE;
    const float* yb = y + b * BATCH_STRIDE;

    // Cooperative halo load of z = x - y (zero-padded outside the volume).
    for (int i = tid; i < HSZ; i += 256) {
        const int zw = i % HW;
        const int t2 = i / HW;
        const int zh = t2 % HH;
        const int zd = t2 / HH;
        const int gd = d0 + zd - 1;
        const int gh = h0 + zh - 1;
        const int gw = w0 + zw - 1;
        float v = 0.0f;
        if ((unsigned)gd < DIM_D && (unsigned)gh < DIM_H && (unsigned)gw < DIM_W) {
            const int gidx = (gd * DIM_H + gh) * DIM_W + gw;
            v = xb[gidx] - yb[gidx];
        }
        tile[i] = v;
    }
    __syncthreads();

    // Each thread owns one (h,w) column, TD voxels deep.
    const int lw = tid & 31;
    const int lh = tid >> 5;
    float partial = 0.0f;

#pragma unroll
    for (int dd = 0; dd < TD; ++dd) {
        const int ci = (dd + 1) * SD + (lh + 1) * SH + (lw + 1);
        const float cval = tile[ci];
#pragma unroll
        for (int kd = 0; kd < 3; ++kd)
#pragma unroll
            for (int kh = 0; kh < 3; ++kh)
#pragma unroll
                for (int kw = 0; kw < 3; ++kw) {
                    const float n = tile[ci + (kd - 1) * SD + (kh - 1) * SH + (kw - 1)];
                    partial += fabsf(cval - n);
                }
    }

    // Wave reduce (WMMA), then block reduce across the 8 waves.
    const float wtot = wave_reduce_wmma(partial);
    const int wave = tid >> 5;
    if ((tid & 31) == 0) wsum[wave] = wtot;
    __syncthreads();
    if (tid == 0) {
        float s = 0.0f;
#pragma unroll
        for (int i = 0; i < 8; ++i) s += wsum[i];
        block_sums[bid] = s;
    }
}

__global__ void __launch_bounds__(256)
cdl_pass2(const float* __restrict__ block_sums, float* __restrict__ out) {
    __shared__ float wsum[8];
    const int tid = threadIdx.x;
    // 1024 partials -> 4 per thread (deterministic, no atomics)
    float p = block_sums[tid] + block_sums[tid + 256] +
              block_sums[tid + 512] + block_sums[tid + 768];
    const float wtot = wave_reduce_wmma(p);
    const int wave = tid >> 5;
    if ((tid & 31) == 0) wsum[wave] = wtot;
    __syncthreads();
    if (tid == 0) {
        float s = 0.0f;
#pragma unroll
        for (int i = 0; i < 8; ++i) s += wsum[i];
        out[0] = s * (1.0f / TOTAL_TERMS);
    }
}

extern "C" void kernel_launch(void* const* d_in, const int* in_sizes, int n_in,
                              void* d_out, int out_size, void* d_ws, size_t ws_size,
                              hipStream_t stream) {
    (void)in_sizes; (void)n_in; (void)out_size; (void)ws_size;
    const float* x = (const float*)d_in[0];
    const float* y = (const float*)d_in[1];
    float* out = (float*)d_out;
    float* ws  = (float*)d_ws;   // needs 1024 floats

    hipLaunchKernelGGL(cdl_pass1, dim3(1024), dim3(256), 0, stream, x, y, ws);
    hipLaunchKernelGGL(cdl_pass2, dim3(1), dim3(256), 0, stream, ws, out);
}